// FragNetLayerA_71064528879626
// MI455X (gfx1250) — compile-verified
//
#include <hip/hip_runtime.h>
#include <hip/hip_bf16.h>
#include <math.h>

typedef float v2f __attribute__((ext_vector_type(2)));
typedef float v8f __attribute__((ext_vector_type(8)));

#define AT_ADD(p, v) __hip_atomic_fetch_add((p), (v), __ATOMIC_RELAXED, __HIP_MEMORY_SCOPE_AGENT)
#define AT_MAX(p, v) __hip_atomic_fetch_max((p), (v), __ATOMIC_RELAXED, __HIP_MEMORY_SCOPE_AGENT)

// ---------------------------------------------------------------------------
// Y[m][n] = sum_k X[m][k] * W[n][k] + b[n]   (X: rows x 256, W: 256 x 256)
// One wave per 16x16 output tile, K stepped by 4 via V_WMMA_F32_16X16X4_F32.
// A layout (16x4 f32): lanes 0-15 -> rows, v0/v1 = K,K+1 ; lanes 16-31 = K+2,K+3
// B layout (4x16 f32): lanes 0-15 -> cols, v0/v1 = K,K+1 ; lanes 16-31 = K+2,K+3
// C/D layout: VGPR i, lanes 0-15 -> (M=i, N=lane); lanes 16-31 -> (M=i+8, N=lane-16)
// ---------------------------------------------------------------------------
__global__ __launch_bounds__(128) void gemm_xwT_bias(
    const float* __restrict__ X, const float* __restrict__ W,
    const float* __restrict__ b, float* __restrict__ Y,
    int rows, int tiles) {
  const int lane = threadIdx.x & 31;
  const int wave = threadIdx.x >> 5;
  const int tile = blockIdx.x * 4 + wave;
  if (tile >= tiles) return;
  const int rowBase = (tile >> 4) << 4;   // 16 col-tiles across D=256
  const int colBase = (tile & 15) << 4;
  if (rowBase >= rows) return;
  const int half = lane >> 4;             // 0: lanes 0-15, 1: lanes 16-31
  const int l = lane & 15;

  v8f acc;
  const float bv = b[colBase + l];
#pragma unroll
  for (int i = 0; i < 8; ++i) acc[i] = bv;   // D = A*B + C, C preloaded w/ bias

  const float* __restrict__ xrow = X + (size_t)(rowBase + l) * 256;
  const float* __restrict__ wrow = W + (size_t)(colBase + l) * 256;

#pragma unroll 4
  for (int k = 0; k < 256; k += 4) {
    v2f av, bw;
    av.x = xrow[k + half * 2 + 0];
    av.y = xrow[k + half * 2 + 1];
    bw.x = wrow[k + half * 2 + 0];   // B[k][n] = W[n][k]
    bw.y = wrow[k + half * 2 + 1];
    acc = __builtin_amdgcn_wmma_f32_16x16x4_f32(
        /*neg_a=*/false, av, /*neg_b=*/false, bw,
        /*c_mod=*/(short)0, acc, /*reuse_a=*/false, /*reuse_b=*/false);
  }

  float* __restrict__ yb = Y + (size_t)rowBase * 256 + colBase;
#pragma unroll
  for (int i = 0; i < 8; ++i) {
    const int m = i + half * 8;
    yb[(size_t)m * 256 + l] = acc[i];
  }
}

// ea[e][j] = sum_k attr[e][k] * Wea[j][k] + bea[j]   (j < 64, K small: 1 or 8)
__global__ __launch_bounds__(256) void ea_proj(
    const float* __restrict__ attr, const float* __restrict__ Wea,
    const float* __restrict__ bea, float* __restrict__ ea, int E, int K) {
  int idx = blockIdx.x * 256 + threadIdx.x;
  if (idx >= E * 64) return;
  const int e = idx >> 6, j = idx & 63;
  float s = bea[j];
  const float* __restrict__ ar = attr + (size_t)e * K;
  const float* __restrict__ wr = Wea + (size_t)j * K;
  for (int k = 0; k < K; ++k) s += ar[k] * wr[k];
  ea[idx] = s;
}

// alpha_t[n,h] = dot64(h[n,h,:], wt[h]);  alpha_s likewise with ws
__global__ __launch_bounds__(256) void node_alphas(
    const float* __restrict__ hbuf, const float* __restrict__ aW,
    int wRowLen, int wsOff, float* __restrict__ at, float* __restrict__ as_,
    int N) {
  int idx = blockIdx.x * 256 + threadIdx.x;   // n*4 + h
  if (idx >= N * 4) return;
  const int h = idx & 3;
  const float* __restrict__ hr = hbuf + (size_t)idx * 64;  // (n*4+h)*64 == n*256+h*64
  const float* __restrict__ wt = aW + (size_t)h * wRowLen;
  const float* __restrict__ wS = wt + wsOff;
  float st = 0.f, ss = 0.f;
#pragma unroll 8
  for (int k = 0; k < 64; ++k) { const float hv = hr[k]; st += hv * wt[k]; ss += hv * wS[k]; }
  at[idx] = st;
  as_[idx] = ss;
}

__global__ __launch_bounds__(256) void fill_val(float* __restrict__ p, float v, int n) {
  int i = blockIdx.x * 256 + threadIdx.x;
  if (i < n) p[i] = v;
}

// logits[e,h] = leakyrelu(at[tgt,h] + as[src,h] + ea[e]·we[h]); atomic max into m[tgt,h]
__global__ __launch_bounds__(256) void edge_logits(
    const int* __restrict__ tgtIdx, const int* __restrict__ srcIdx,
    const float* __restrict__ ea, int de,
    const float* __restrict__ aW, int wRowLen,
    const float* __restrict__ at, const float* __restrict__ as_,
    float* __restrict__ logits, float* __restrict__ m, int E, int nIdx) {
  int idx = blockIdx.x * 256 + threadIdx.x;   // e*4 + h
  if (idx >= E * 4) return;
  const int e = idx >> 2, h = idx & 3;
  int tgt, src;
  float edot = 0.f;
  if (e < nIdx) {
    tgt = tgtIdx[e];
    src = srcIdx[e];
    const float* __restrict__ ear = ea + (size_t)e * de;
    const float* __restrict__ we = aW + (size_t)h * wRowLen + 64;
#pragma unroll 8
    for (int k = 0; k < de; ++k) edot += ear[k] * we[k];
  } else {
    tgt = src = e - nIdx;   // self-loop edges (atom graph); ea rows are zero
  }
  float l = at[tgt * 4 + h] + as_[src * 4 + h] + edot;
  l = l > 0.f ? l : 0.2f * l;   // leaky_relu(0.2)
  logits[idx] = l;
  AT_MAX(&m[tgt * 4 + h], l);
}

// logits[e,h] <- exp(logits - m[tgt,h]); atomic add into s[tgt,h]
__global__ __launch_bounds__(256) void edge_exp(
    const int* __restrict__ tgtIdx, float* __restrict__ logits,
    const float* __restrict__ m, float* __restrict__ s, int E, int nIdx) {
  int idx = blockIdx.x * 256 + threadIdx.x;
  if (idx >= E * 4) return;
  const int e = idx >> 2, h = idx & 3;
  const int tgt = (e < nIdx) ? tgtIdx[e] : (e - nIdx);
  const float ex = __expf(logits[idx] - m[tgt * 4 + h]);
  logits[idx] = ex;
  AT_ADD(&s[tgt * 4 + h], ex);
}

// out[tgt, c] += (exp[e,h]/s[tgt,h]) * h[src, c],  c in [0,256), h = c/64
__global__ __launch_bounds__(256) void edge_scatter(
    const int* __restrict__ tgtIdx, const int* __restrict__ srcIdx,
    const float* __restrict__ logits, const float* __restrict__ s,
    const float* __restrict__ hbuf, float* __restrict__ out, int E, int nIdx) {
  const int e = blockIdx.x;
  const int c = threadIdx.x;
  const int h = c >> 6;
  int tgt, src;
  if (e < nIdx) { tgt = tgtIdx[e]; src = srcIdx[e]; }
  else          { tgt = src = e - nIdx; }
  const float p = logits[e * 4 + h] / s[tgt * 4 + h];
  AT_ADD(&out[(size_t)tgt * 256 + c], p * hbuf[(size_t)src * 256 + c]);
}

// agg[frag[a], c] += x[a, c]
__global__ __launch_bounds__(256) void frag_agg(
    const int* __restrict__ a2f, const float* __restrict__ x,
    float* __restrict__ agg, int NA_) {
  int idx = blockIdx.x * 256 + threadIdx.x;
  if (idx >= NA_ * 256) return;
  const int a = idx >> 8, c = idx & 255;
  AT_ADD(&agg[(size_t)a2f[a] * 256 + c], x[idx]);
}

// ---------------------------------------------------------------------------
extern "C" void kernel_launch(void* const* d_in, const int* in_sizes, int n_in,
                              void* d_out, int out_size, void* d_ws, size_t ws_size,
                              hipStream_t stream) {
  const int D_ = 256, H_ = 4;
  const int NA   = in_sizes[0] / D_;
  const int EB   = in_sizes[1] / 2;
  const int EF   = in_sizes[3] / 2;
  const int NF   = in_sizes[4] / D_;
  const int EBG  = in_sizes[7] / 2;
  const int EFBG = in_sizes[10] / 2;

  const float* x_atoms    = (const float*)d_in[0];
  const int*   edge_index = (const int*)d_in[1];
  const int*   frag_index = (const int*)d_in[3];
  const int*   a2f        = (const int*)d_in[5];
  const float* nf_bond    = (const float*)d_in[6];
  const int*   idx_bond   = (const int*)d_in[7];
  const float* attr_bond  = (const float*)d_in[8];
  const float* nf_fbond   = (const float*)d_in[9];
  const int*   idx_fbond  = (const int*)d_in[10];
  const float* attr_fbond = (const float*)d_in[11];
  const float* W_pb  = (const float*)d_in[12]; const float* b_pb  = (const float*)d_in[13];
  const float* W_eab = (const float*)d_in[14]; const float* b_eab = (const float*)d_in[15];
  const float* a_b   = (const float*)d_in[16];
  const float* W_pa  = (const float*)d_in[17]; const float* b_pa  = (const float*)d_in[18];
  const float* a_w   = (const float*)d_in[19];
  const float* W_pfb = (const float*)d_in[20]; const float* b_pfb = (const float*)d_in[21];
  const float* W_eafb = (const float*)d_in[22]; const float* b_eafb = (const float*)d_in[23];
  const float* f_a_b = (const float*)d_in[24];
  const float* f_w   = (const float*)d_in[25];

  float* out = (float*)d_out;
  const size_t oFrags = (size_t)NA * D_;
  const size_t oBond  = oFrags + (size_t)NF * D_;
  const size_t oFbond = oBond + (size_t)EB * D_;

  // workspace bump allocation (floats)
  const size_t maxN = (size_t)EB;   // >= NA, EF, NF
  const size_t maxE = (size_t)EBG;  // >= EB+NA, EFBG, EF
  float* ws    = (float*)d_ws;
  float* bufH  = ws; ws += (size_t)EB * D_;     // projected node features
  float* bufEA = ws; ws += (size_t)EBG * 64;    // projected edge attrs
  float* bufLG = ws; ws += maxE * H_;           // logits -> exp values
  float* bufM  = ws; ws += maxN * H_;           // segment max
  float* bufS  = ws; ws += maxN * H_;           // segment sum
  float* bufAT = ws; ws += maxN * H_;           // alpha_t
  float* bufAS = ws; ws += maxN * H_;           // alpha_s
  float* bufFR = ws; ws += (size_t)NF * D_;     // x_frags_agg

  hipMemsetAsync(d_out, 0, (size_t)out_size * sizeof(float), stream);

  auto run_gat = [&](const float* hbuf, int N, int E, int nIdx,
                     const int* tgt, const int* src,
                     const float* ea, int de,
                     const float* aW, int wRowLen, float* outp) {
    const int nh = N * H_;
    const int eh = E * H_;
    node_alphas<<<(nh + 255) / 256, 256, 0, stream>>>(hbuf, aW, wRowLen, 64 + de,
                                                      bufAT, bufAS, N);
    fill_val<<<(nh + 255) / 256, 256, 0, stream>>>(bufM, -INFINITY, nh);
    hipMemsetAsync(bufS, 0, (size_t)nh * sizeof(float), stream);
    edge_logits<<<(eh + 255) / 256, 256, 0, stream>>>(tgt, src, ea, de, aW, wRowLen,
                                                      bufAT, bufAS, bufLG, bufM, E, nIdx);
    edge_exp<<<(eh + 255) / 256, 256, 0, stream>>>(tgt, bufLG, bufM, bufS, E, nIdx);
    edge_scatter<<<E, 256, 0, stream>>>(tgt, src, bufLG, bufS, hbuf, outp, E, nIdx);
  };

  // ---- Stage 1: bond graph GAT -> new_bond_features ----
  {
    const int tiles = EB;   // (EB/16) row tiles * 16 col tiles
    gemm_xwT_bias<<<(tiles + 3) / 4, 128, 0, stream>>>(nf_bond, W_pb, b_pb, bufH, EB, tiles);
    const int n = EBG * 64;
    ea_proj<<<(n + 255) / 256, 256, 0, stream>>>(attr_bond, W_eab, b_eab, bufEA, EBG, 1);
    run_gat(bufH, EB, EBG, EBG, idx_bond, idx_bond + EBG, bufEA, 64, a_b, 192, out + oBond);
  }

  // ---- Stage 2: atom graph GAT (with self-loops) -> x_atoms_new ----
  {
    const int tiles = NA;
    gemm_xwT_bias<<<(tiles + 3) / 4, 128, 0, stream>>>(x_atoms, W_pa, b_pa, bufH, NA, tiles);
    // src = row0, tgt = row1; loop edges e >= EB have src=tgt=e-EB, ea contribution 0
    run_gat(bufH, NA, EB + NA, EB, edge_index + EB, edge_index, out + oBond, 256,
            a_w, 384, out /* oAtoms = 0 */);
  }

  // ---- Stage 3: fragment aggregation ----
  hipMemsetAsync(bufFR, 0, (size_t)NF * D_ * sizeof(float), stream);
  frag_agg<<<NA, 256, 0, stream>>>(a2f, out, bufFR, NA);

  // ---- Stage 4: fbond graph GAT -> new_fbond_features ----
  {
    const int tiles = EF;
    gemm_xwT_bias<<<(tiles + 3) / 4, 128, 0, stream>>>(nf_fbond, W_pfb, b_pfb, bufH, EF, tiles);
    const int n = EFBG * 64;
    ea_proj<<<(n + 255) / 256, 256, 0, stream>>>(attr_fbond, W_eafb, b_eafb, bufEA, EFBG, 8);
    run_gat(bufH, EF, EFBG, EFBG, idx_fbond, idx_fbond + EFBG, bufEA, 64, f_a_b, 192,
            out + oFbond);
  }

  // ---- Stage 5: frag graph GAT -> x_frags_new ----
  // src = frag_index[0], tgt = frag_index[1]; h = x_frags_agg (no projection)
  run_gat(bufFR, NF, EF, EF, frag_index + EF, frag_index, out + oFbond, 256,
          f_w, 384, out + oFrags);
}